// DirectionalPropagation_11845519803030
// MI455X (gfx1250) — compile-verified
//
#include <hip/hip_runtime.h>
#include <hip/hip_bf16.h>
#include <math.h>

typedef __attribute__((ext_vector_type(2))) float v2f;
typedef __attribute__((ext_vector_type(8))) float v8f;

#define HIDDEN 128
#define TRANS  64
#define KHOPS  3

// ---------------------------------------------------------------------------
// Kernel 1: XP[N,128] = x[N,128] @ [W_trans[0:128,:] | W_trans[128:256,:]]
// fp32 WMMA 16x16x4, one 16x16 output tile per wave, 32 K-steps.
// grid = (ceil(N/16), 2), block = 128 (4 waves); wave w of blockIdx.y group
// handles column tile (blockIdx.y*4 + w).
// ---------------------------------------------------------------------------
__global__ __launch_bounds__(128)
void xp_gemm_wmma(const float* __restrict__ x, const float* __restrict__ Wt,
                  float* __restrict__ XP, int nRows) {
  const int wave = threadIdx.x >> 5;
  const int lane = threadIdx.x & 31;
  const int r    = lane & 15;   // row (A) / col (B,D) within tile
  const int kh   = lane >> 4;   // K half-select (ISA 32-bit A 16x4 layout)

  const int tileN = blockIdx.y * 4 + wave;      // 0..7
  const int n0    = tileN * 16;                 // output column base
  const int m0    = blockIdx.x * 16;            // output row base

  // Column j<64 comes from W_trans rows [0,128) ("src" half),
  // j>=64 from rows [128,256) ("dst" half). A 16-wide tile never straddles.
  const float* Wbase = Wt + ((n0 >= TRANS) ? (size_t)HIDDEN * TRANS : 0);
  const int c0 = n0 & (TRANS - 1);

  const int rr = min(m0 + r, nRows - 1);        // clamp (N is 16-aligned anyway)
  const float* arow = x + (size_t)rr * HIDDEN;

  v8f acc = {};
  #pragma unroll 4
  for (int k = 0; k < HIDDEN; k += 4) {
    const int ka = k + 2 * kh;                  // this half-wave's K pair
    v2f a, b;
    a.x = arow[ka];
    a.y = arow[ka + 1];
    b.x = Wbase[(size_t)ka * TRANS + c0 + r];
    b.y = Wbase[(size_t)(ka + 1) * TRANS + c0 + r];
    acc = __builtin_amdgcn_wmma_f32_16x16x4_f32(
        /*neg_a=*/false, a, /*neg_b=*/false, b,
        /*c_mod=*/(short)0, acc, /*reuse_a=*/false, /*reuse_b=*/false);
  }

  // D layout: VGPR j -> row (m0 + j + 8*kh), col (n0 + r).
  float* outp = XP + (size_t)(m0 + 8 * kh) * HIDDEN + n0 + r;
  if (m0 + 16 <= nRows) {
    // full tile: unguarded stores, no exec-mask branching
    #pragma unroll
    for (int j = 0; j < 8; ++j)
      outp[(size_t)j * HIDDEN] = acc[j];
  } else {
    #pragma unroll
    for (int j = 0; j < 8; ++j) {
      if (m0 + 8 * kh + j < nRows) outp[(size_t)j * HIDDEN] = acc[j];
    }
  }
}

// ---------------------------------------------------------------------------
// Kernel 2: per-edge sigmoid weight. One wave (32 lanes) per edge.
// ew[e] = sigmoid( sum_k Wm[attr+k]*relu(XP[src,k]+XP[dst,64+k]+b_trans[k])
//                  + sum_a Wm[a]*eattr[e,a] + bm )
// ---------------------------------------------------------------------------
__global__ __launch_bounds__(256)
void edge_ew_kernel(const float* __restrict__ XP, const int* __restrict__ ei,
                    const float* __restrict__ eattr, int attrDim,
                    const float* __restrict__ btrans,
                    const float* __restrict__ Wm, const float* __restrict__ bm,
                    float* __restrict__ ew, int E) {
  const int lane = threadIdx.x & 31;
  const int e = blockIdx.x * (blockDim.x >> 5) + (threadIdx.x >> 5);
  if (e >= E) return;

  const int src = ei[e];
  const int dst = ei[E + e];
  const float* xs = XP + (size_t)src * HIDDEN;         // src projection
  const float* xd = XP + (size_t)dst * HIDDEN + TRANS; // dst projection

  float acc = 0.f;
  #pragma unroll
  for (int t = 0; t < 2; ++t) {
    const int k = lane + 32 * t;
    float v = xs[k] + xd[k] + btrans[k];
    v = fmaxf(v, 0.f);                                  // relu
    acc += v * Wm[attrDim + k];
  }
  // wave32 butterfly reduction
  #pragma unroll
  for (int off = 16; off > 0; off >>= 1)
    acc += __shfl_xor(acc, off, 32);

  if (lane == 0) {
    float z = acc + bm[0];
    for (int a = 0; a < attrDim; ++a)
      z += eattr[(size_t)e * attrDim + a] * Wm[a];
    ew[e] = 1.f / (1.f + __expf(-z));
  }
}

// ---------------------------------------------------------------------------
// Propagation helpers. All mask values are >= 0, so float-max == int-max on
// the raw bit pattern (deterministic regardless of atomic ordering).
// ---------------------------------------------------------------------------
__global__ void copy_f32(const float* __restrict__ a, float* __restrict__ b, int n) {
  int i = blockIdx.x * blockDim.x + threadIdx.x;
  if (i < n) b[i] = a[i];
}

__global__ void hop_kernel(const int* __restrict__ ei, const float* __restrict__ ew,
                           const float* __restrict__ mcur, float* __restrict__ mnext,
                           int E) {
  int e = blockIdx.x * blockDim.x + threadIdx.x;
  if (e >= E) return;
  const int src = ei[e];
  const int dst = ei[E + e];
  const float v = mcur[src] * ew[e];
  atomicMax((int*)(mnext + dst), __float_as_int(v));
}

__global__ void final_max(const float* __restrict__ mask,
                          const float* __restrict__ mpos,
                          const float* __restrict__ mdom,
                          float* __restrict__ out, int n) {
  int i = blockIdx.x * blockDim.x + threadIdx.x;
  if (i < n) out[i] = fmaxf(mask[i], fmaxf(mpos[i], mdom[i]));
}

// ---------------------------------------------------------------------------
extern "C" void kernel_launch(void* const* d_in, const int* in_sizes, int n_in,
                              void* d_out, int out_size, void* d_ws, size_t ws_size,
                              hipStream_t stream) {
  const float* x        = (const float*)d_in[0];
  const int*   dom_ei   = (const int*)  d_in[1];
  const int*   sp_ei    = (const int*)  d_in[2];
  const float* dom_attr = (const float*)d_in[3];
  const float* sp_attr  = (const float*)d_in[4];
  const float* mask     = (const float*)d_in[5];
  const float* W_trans  = (const float*)d_in[6];
  const float* b_trans  = (const float*)d_in[7];
  const float* W_dom    = (const float*)d_in[8];
  const float* b_dom    = (const float*)d_in[9];
  const float* W_pos    = (const float*)d_in[10];
  const float* b_pos    = (const float*)d_in[11];
  float* out = (float*)d_out;

  const int N = in_sizes[5];          // mask is [N,1]
  const int E = in_sizes[1] / 2;      // edge_index is [2,E]

  // Workspace layout (floats)
  float* XP     = (float*)d_ws;                  // N*128
  float* ew_pos = XP + (size_t)N * HIDDEN;       // E
  float* ew_dom = ew_pos + E;                    // E
  float* mA     = ew_dom + E;                    // N
  float* mB     = mA + N;                        // N
  float* mC     = mB + N;                        // N
  float* mD     = mC + N;                        // N

  // 1) Node-side projection GEMM via fp32 WMMA
  dim3 ggrid((N + 15) / 16, 2);
  xp_gemm_wmma<<<ggrid, 128, 0, stream>>>(x, W_trans, XP, N);

  // 2) Per-edge sigmoid weights (one wave per edge)
  const int wavesPerBlk = 256 / 32;
  const int eblocks = (E + wavesPerBlk - 1) / wavesPerBlk;
  edge_ew_kernel<<<eblocks, 256, 0, stream>>>(XP, sp_ei,  sp_attr,  4, b_trans, W_pos, b_pos, ew_pos, E);
  edge_ew_kernel<<<eblocks, 256, 0, stream>>>(XP, dom_ei, dom_attr, 1, b_trans, W_dom, b_dom, ew_dom, E);

  // 3) K hops of masked max-propagation, double-buffered, per branch
  const int nblk = (N + 255) / 256;
  const int hblk = (E + 255) / 256;

  // spatial branch: mA <-> mB
  copy_f32<<<nblk, 256, 0, stream>>>(mask, mA, N);
  {
    float* cur = mA; float* nxt = mB;
    for (int h = 0; h < KHOPS; ++h) {
      copy_f32<<<nblk, 256, 0, stream>>>(cur, nxt, N);
      hop_kernel<<<hblk, 256, 0, stream>>>(sp_ei, ew_pos, cur, nxt, E);
      float* t = cur; cur = nxt; nxt = t;
    }
    if (cur != mA) copy_f32<<<nblk, 256, 0, stream>>>(cur, mA, N); // result in mA
  }

  // dom branch: mC <-> mD
  copy_f32<<<nblk, 256, 0, stream>>>(mask, mC, N);
  {
    float* cur = mC; float* nxt = mD;
    for (int h = 0; h < KHOPS; ++h) {
      copy_f32<<<nblk, 256, 0, stream>>>(cur, nxt, N);
      hop_kernel<<<hblk, 256, 0, stream>>>(dom_ei, ew_dom, cur, nxt, E);
      float* t = cur; cur = nxt; nxt = t;
    }
    if (cur != mC) copy_f32<<<nblk, 256, 0, stream>>>(cur, mC, N); // result in mC
  }

  // 4) out = max(mask, pos_mask, dom_mask)
  final_max<<<nblk, 256, 0, stream>>>(mask, mA, mC, out, N);
}